// DiscoMixerBlock_26869315403889
// MI455X (gfx1250) — compile-verified
//
#include <hip/hip_runtime.h>
#include <hip/hip_bf16.h>
#include <math.h>
#include <stdint.h>

// ---------------- problem dims (match reference) ----------------
#define H_    90
#define W_    180
#define P_    (H_ * W_)   // 16200
#define MD_   130
#define L_    128
#define NH_   8
#define HD_   16
#define S_    16
#define K_    25
#define NNB_  25
#define MH_   32
#define FH_   256

typedef float v2f  __attribute__((ext_vector_type(2)));
typedef float v8f  __attribute__((ext_vector_type(8)));
typedef unsigned int u32x4 __attribute__((ext_vector_type(4)));
typedef int   i32x4 __attribute__((ext_vector_type(4)));
typedef int   i32x8 __attribute__((ext_vector_type(8)));

__device__ __forceinline__ float gelu_exact(float x) {
  return 0.5f * x * (1.0f + erff(x * 0.70710678118654752440f));
}

__device__ __forceinline__ v8f zero8() {
  v8f z = {0.f, 0.f, 0.f, 0.f, 0.f, 0.f, 0.f, 0.f};
  return z;
}

// Full-precision CDNA5 WMMA: D(16x16,f32) = A(16x4,f32) * B(4x16,f32) + C
__device__ __forceinline__ v8f wmma_f32(v2f a, v2f b, v8f c) {
  return __builtin_amdgcn_wmma_f32_16x16x4_f32(false, a, false, b,
                                               (short)0, c, false, false);
}

// ---------------- Async Global->LDS (GLOBAL_LOAD_ASYNC_TO_LDS_B64) ----------
// Writes 8 bytes directly to LDS, tracked by ASYNCcnt. lds_addr in VDST VGPR.
__device__ __forceinline__ void async_load_lds_b64(unsigned lds_addr,
                                                   unsigned long long gaddr) {
  asm volatile("global_load_async_to_lds_b64 %0, %1, off"
               :: "v"(lds_addr), "v"(gaddr) : "memory");
}
__device__ __forceinline__ void wait_asynccnt0() {
  asm volatile("s_wait_asynccnt 0" ::: "memory");
}

// ---------------- Tensor Data Mover: 2D tile Global -> LDS ----------------
// Builds a D# per CDNA5 ISA and issues TENSOR_LOAD_TO_LDS.
// data_size = 4 bytes. Tracked by TENSORcnt.
__device__ __forceinline__ void tdm_load_2d_f32(unsigned lds_off, const void* gptr,
                                                unsigned tensor_d0, unsigned tensor_d1,
                                                unsigned tile_d0, unsigned tile_d1,
                                                unsigned stride0 /* elements */) {
  unsigned long long ga = (unsigned long long)(uintptr_t)gptr;
  u32x4 g0;
  g0[0] = 1u;                                         // count=1 (valid user D#)
  g0[1] = lds_off;                                    // lds_addr [63:32]
  g0[2] = (unsigned)(ga & 0xFFFFFFFFu);               // global_addr lo
  g0[3] = (unsigned)((ga >> 32) & 0x01FFFFFFu)        // global_addr hi
          | 0x80000000u;                              // type=2 ("image")
  i32x8 g1;
  g1[0] = 0x00020000;                                 // data_size=2 (4B)
  g1[1] = (int)((tensor_d0 & 0xFFFFu) << 16);         // tensor_dim0 [79:48] lo16
  g1[2] = (int)((tensor_d0 >> 16) | ((tensor_d1 & 0xFFFFu) << 16)); // d0 hi | dim1 lo
  g1[3] = (int)((tensor_d1 >> 16) | (tile_d0 << 16)); // dim1 hi | tile_dim0 [127:112]
  g1[4] = (int)(tile_d1 & 0xFFFFu);                   // tile_dim1; tile_dim2=0 (2D)
  g1[5] = (int)stride0;                               // tensor_dim0_stride lo32
  g1[6] = 0;
  g1[7] = 0;
  i32x4 z4 = {0, 0, 0, 0};
#if defined(__clang_major__) && (__clang_major__ >= 23)
  i32x8 z8 = {0, 0, 0, 0, 0, 0, 0, 0};
  __builtin_amdgcn_tensor_load_to_lds(g0, g1, z4, z4, z8, 0);
#else
  __builtin_amdgcn_tensor_load_to_lds(g0, g1, z4, z4, 0);
#endif
}

// =====================================================================
// Kernel 1: per-point gather + disco contraction + per-head MLP.
// One 128-thread block (4 wave32) per point.
//   W2[n][s] = sum_k disco_w[s][k] * basis[k][p][n]      (25x16, n padded to 28/32)
//   Y[s][c]  = sum_n W2[n][s] * G[n][c] + disco_b[s]     (WMMA 16x16x4, K=28)
//   h1       = gelu(hfeat(16x16) * head_w1[h](16x32))    (WMMA, K=16)
//   x_mix[c] = sum_m h1[d][m]*head_w2[h][m] + b2[h] + x_res[c]
// Writes x_mix into out[:, :128] and passes sin/cos through out[:, 128:130].
// =====================================================================
__global__ __launch_bounds__(128)
void disco_head_kernel(const float* __restrict__ x,
                       const int*   __restrict__ nbr_idx,
                       const float* __restrict__ basis,
                       const float* __restrict__ disco_w,
                       const float* __restrict__ disco_b,
                       const float* __restrict__ head_w1,
                       const float* __restrict__ head_b1,
                       const float* __restrict__ head_w2,
                       const float* __restrict__ head_b2,
                       float* __restrict__ out) {
  __shared__ float sBas[K_][NNB_];        // basis at point p      (25x25)
  __shared__ float sDW[S_][K_];           // disco_w               (16x25)
  __shared__ float sDB[S_];
  __shared__ float sW2[32][S_];           // fused weights, [n][s], n padded
  __shared__ float sG[28][L_];            // gathered neighbors, K padded to 28
  __shared__ float sY[S_][L_];            // disco output (s, c)
  __shared__ float sHW1[NH_][S_][MH_];    // head_w1               (8x16x32)
  __shared__ float sH1[NH_][HD_][MH_];    // gelu'd hidden
  __shared__ int   sIdx[NNB_];

  const int p    = blockIdx.x;
  const int tid  = threadIdx.x;
  const int lane = tid & 31;
  const int wave = tid >> 5;
  const int m    = lane & 15;          // matrix row/col within fragment
  const int kb   = (lane >> 4) << 1;   // K sub-offset (0 or 2) for A/B frags
  const int rb   = (lane >> 4) << 3;   // D row base (0 or 8)

  // ---- stage 0: stage small operands into LDS ----
  if (tid < NNB_) sIdx[tid] = nbr_idx[p * NNB_ + tid];
  for (int i = tid; i < K_ * NNB_; i += 128) {
    int k = i / NNB_, n = i % NNB_;
    sBas[k][n] = basis[(size_t)k * (P_ * NNB_) + (size_t)p * NNB_ + n];
  }
  for (int i = tid; i < S_ * K_; i += 128) sDW[i / K_][i % K_] = disco_w[i];
  if (tid < S_) sDB[tid] = disco_b[tid];
  for (int i = tid; i < NH_ * S_ * MH_; i += 128) (&sHW1[0][0][0])[i] = head_w1[i];
  __syncthreads();

  // ---- stage 1a: async gather of 25 neighbor rows directly into LDS ----
  // Rows are 520 B apart (only 8B-aligned) -> B64 transfers; 64 chunks/row.
  for (int i = tid; i < NNB_ * 64; i += 128) {
    int n = i >> 6, ch = i & 63;
    int row = sIdx[n];
    unsigned long long ga =
        (unsigned long long)(uintptr_t)(x + (size_t)row * MD_ + ch * 2);
    unsigned la = (unsigned)(uintptr_t)&sG[n][ch * 2];
    async_load_lds_b64(la, ga);
  }
  for (int i = tid; i < 3 * L_; i += 128) sG[NNB_ + (i >> 7)][i & (L_ - 1)] = 0.f;

  // ---- stage 1b: W2[n][s] (tiny per-point matmul, VALU; overlaps gather) ----
  for (int i = tid; i < 32 * S_; i += 128) {
    int n = i >> 4, s = i & 15;
    float acc = 0.f;
    if (n < NNB_) {
      #pragma unroll
      for (int k = 0; k < K_; ++k) acc += sDW[s][k] * sBas[k][n];
    }
    sW2[n][s] = acc;
  }
  wait_asynccnt0();
  __syncthreads();

  // ---- stage 2: Y(16x128) = W2^T(16x28) x G(28x128) via WMMA f32 ----
  // K outermost: A fragment loaded once, fed to both c-tiles of this wave.
  {
    v8f acc[2] = {zero8(), zero8()};
    #pragma unroll
    for (int kk = 0; kk < 7; ++kk) {  // K = 28 in steps of 4
      v2f a;
      a.x = sW2[kk * 4 + kb + 0][m];          // A[s][n] = W2[n][s]
      a.y = sW2[kk * 4 + kb + 1][m];
      #pragma unroll
      for (int t = 0; t < 2; ++t) {
        const int ct = wave * 2 + t;          // c-tile 0..7
        v2f b;
        b.x = sG[kk * 4 + kb + 0][ct * 16 + m]; // B[n][c]
        b.y = sG[kk * 4 + kb + 1][ct * 16 + m];
        acc[t] = wmma_f32(a, b, acc[t]);
      }
    }
    #pragma unroll
    for (int t = 0; t < 2; ++t) {
      const int ct = wave * 2 + t;
      #pragma unroll
      for (int r = 0; r < 8; ++r)
        sY[rb + r][ct * 16 + m] = acc[t][r] + sDB[rb + r];
    }
  }
  __syncthreads();

  // ---- stage 3: per-head h1 = gelu(hfeat * w1 + b1), hfeat[d][s] = Y[s][h*16+d] ----
  for (int hh = 0; hh < 2; ++hh) {
    const int h = wave * 2 + hh;              // this wave's heads
    v8f acc[2] = {zero8(), zero8()};
    #pragma unroll
    for (int kk = 0; kk < 4; ++kk) {          // K = 16
      v2f a;
      a.x = sY[kk * 4 + kb + 0][h * 16 + m];  // A[d][s]
      a.y = sY[kk * 4 + kb + 1][h * 16 + m];
      #pragma unroll
      for (int nt = 0; nt < 2; ++nt) {
        v2f b;
        b.x = sHW1[h][kk * 4 + kb + 0][nt * 16 + m];  // B[s][mh]
        b.y = sHW1[h][kk * 4 + kb + 1][nt * 16 + m];
        acc[nt] = wmma_f32(a, b, acc[nt]);
      }
    }
    #pragma unroll
    for (int nt = 0; nt < 2; ++nt) {
      #pragma unroll
      for (int r = 0; r < 8; ++r) {
        int d = rb + r, mm = nt * 16 + m;
        sH1[h][d][mm] = gelu_exact(acc[nt][r] + head_b1[h * MH_ + mm]);
      }
    }
  }
  __syncthreads();

  // ---- stage 4: hout + residual, write x_mix and sin/cos passthrough ----
  {
    const int h = tid >> 4, d = tid & 15;
    float acc = head_b2[h];
    #pragma unroll
    for (int mm = 0; mm < MH_; ++mm) acc += sH1[h][d][mm] * head_w2[h * MH_ + mm];
    const int c = tid;                // == h*16 + d
    out[(size_t)p * MD_ + c] = acc + x[(size_t)p * MD_ + c];
  }
  if (tid < 2)
    out[(size_t)p * MD_ + L_ + tid] = x[(size_t)p * MD_ + L_ + tid];
}

// =====================================================================
// Kernel 2: FFN over 64-point tiles.
//   Hf(64x256)  = gelu(X(64x132pad) * W1(132x256) + b1)
//   Out(64x128) = Hf * W2(256x128) + b2 + x_mix   (residual, from sX)
// Weight chunks streamed to LDS with the CDNA5 Tensor Data Mover.
// Reads x_mix+sincos from `io` (written by kernel 1) and overwrites in place.
// =====================================================================
__global__ __launch_bounds__(128)
void ffn_kernel(const float* __restrict__ ffn_w1,   // (130,256)
                const float* __restrict__ ffn_b1,   // (256)
                const float* __restrict__ ffn_w2,   // (256,128)
                const float* __restrict__ ffn_b2,   // (128)
                float* __restrict__ io) {           // (P,130) in/out
  __shared__ float sX[64][132];        // x_full tile, K padded 130->132
  __shared__ float sW1c[132][64];      // W1 column chunk (rows 130/131 zero)
  __shared__ float sHf[64][FH_ + 4];   // gelu'd hidden (padded: break 256-stride banks)
  __shared__ float sW2c[64][L_];       // W2 row chunk
  __shared__ float sB1[FH_];
  __shared__ float sB2[L_];

  const int tid  = threadIdx.x;
  const int lane = tid & 31;
  const int wave = tid >> 5;
  const int m    = lane & 15;
  const int kb   = (lane >> 4) << 1;
  const int rb   = (lane >> 4) << 3;
  const int rows0 = blockIdx.x * 64;

  // ---- load X tile (cols 0..129 from io; pad cols/rows with zeros) ----
  for (int i = tid; i < 64 * 132; i += 128) {
    int r = i / 132, c = i % 132;
    int row = rows0 + r;
    float v = 0.f;
    if (row < P_ && c < MD_) v = io[(size_t)row * MD_ + c];
    sX[r][c] = v;
  }
  for (int i = tid; i < FH_; i += 128) sB1[i] = ffn_b1[i];
  sB2[tid] = ffn_b2[tid];
  // zero K-pad rows of the W1 chunk once; TDM only ever writes rows 0..129
  for (int i = tid; i < 2 * 64; i += 128) sW1c[130 + (i >> 6)][i & 63] = 0.f;
  __syncthreads();

  // ---- GEMM1 + gelu: 4 column chunks of 64 ----
  for (int nc = 0; nc < 4; ++nc) {
    __syncthreads();  // previous chunk's sW1c reads complete
    if (wave == 0) {
      // W1 tile: 64 cols x 130 rows, row stride 256 elements
      tdm_load_2d_f32((unsigned)(uintptr_t)&sW1c[0][0],
                      ffn_w1 + nc * 64, FH_, MD_, 64, MD_, FH_);
      __builtin_amdgcn_s_wait_tensorcnt(0);
    }
    __syncthreads();  // TDM data visible to all waves

    v8f acc[4] = {zero8(), zero8(), zero8(), zero8()};
    #pragma unroll
    for (int kk = 0; kk < 33; ++kk) {   // K = 132; A fragment shared by 4 N-tiles
      v2f a;
      a.x = sX[wave * 16 + m][kk * 4 + kb + 0];
      a.y = sX[wave * 16 + m][kk * 4 + kb + 1];
      #pragma unroll
      for (int nt = 0; nt < 4; ++nt) {
        v2f b;
        b.x = sW1c[kk * 4 + kb + 0][nt * 16 + m];
        b.y = sW1c[kk * 4 + kb + 1][nt * 16 + m];
        acc[nt] = wmma_f32(a, b, acc[nt]);
      }
    }
    #pragma unroll
    for (int nt = 0; nt < 4; ++nt) {
      #pragma unroll
      for (int r = 0; r < 8; ++r) {
        int col = nc * 64 + nt * 16 + m;
        sHf[wave * 16 + rb + r][col] = gelu_exact(acc[nt][r] + sB1[col]);
      }
    }
  }

  // ---- GEMM2: Out = Hf(64x256) * W2(256x128), K streamed in chunks of 64 ----
  v8f acc2[8];
  #pragma unroll
  for (int i = 0; i < 8; ++i) acc2[i] = zero8();
  for (int kc = 0; kc < 4; ++kc) {
    __syncthreads();  // (kc==0: sHf complete; kc>0: sW2c reads complete)
    if (wave == 0) {
      // W2 tile: full 128-wide rows, 64 rows, contiguous
      tdm_load_2d_f32((unsigned)(uintptr_t)&sW2c[0][0],
                      ffn_w2 + (size_t)kc * 64 * L_, L_, FH_, L_, 64, L_);
      __builtin_amdgcn_s_wait_tensorcnt(0);
    }
    __syncthreads();
    #pragma unroll
    for (int kk = 0; kk < 16; ++kk) {   // K chunk = 64; A shared by 8 N-tiles
      v2f a;
      a.x = sHf[wave * 16 + m][kc * 64 + kk * 4 + kb + 0];
      a.y = sHf[wave * 16 + m][kc * 64 + kk * 4 + kb + 1];
      #pragma unroll
      for (int nt = 0; nt < 8; ++nt) {
        v2f b;
        b.x = sW2c[kk * 4 + kb + 0][nt * 16 + m];
        b.y = sW2c[kk * 4 + kb + 1][nt * 16 + m];
        acc2[nt] = wmma_f32(a, b, acc2[nt]);
      }
    }
  }

  // ---- epilogue: + b2 + residual(x_mix, still held in sX), store ----
  for (int nt = 0; nt < 8; ++nt) {
    #pragma unroll
    for (int r = 0; r < 8; ++r) {
      int lr  = wave * 16 + rb + r;
      int row = rows0 + lr;
      int c   = nt * 16 + m;
      if (row < P_)
        io[(size_t)row * MD_ + c] = acc2[nt][r] + sB2[c] + sX[lr][c];
    }
  }
  // sin/cos columns already written by kernel 1 — left untouched.
}

// =====================================================================
extern "C" void kernel_launch(void* const* d_in, const int* in_sizes, int n_in,
                              void* d_out, int out_size, void* d_ws, size_t ws_size,
                              hipStream_t stream) {
  (void)in_sizes; (void)n_in; (void)out_size; (void)d_ws; (void)ws_size;
  const float* x       = (const float*)d_in[0];
  const int*   nbr     = (const int*)  d_in[1];
  const float* basis   = (const float*)d_in[2];
  const float* disco_w = (const float*)d_in[3];
  const float* disco_b = (const float*)d_in[4];
  const float* head_w1 = (const float*)d_in[5];
  const float* head_b1 = (const float*)d_in[6];
  const float* head_w2 = (const float*)d_in[7];
  const float* head_b2 = (const float*)d_in[8];
  const float* ffn_w1  = (const float*)d_in[9];
  const float* ffn_b1  = (const float*)d_in[10];
  const float* ffn_w2  = (const float*)d_in[11];
  const float* ffn_b2  = (const float*)d_in[12];
  float* out = (float*)d_out;

  disco_head_kernel<<<P_, 128, 0, stream>>>(x, nbr, basis, disco_w, disco_b,
                                            head_w1, head_b1, head_w2, head_b2,
                                            out);
  ffn_kernel<<<(P_ + 63) / 64, 128, 0, stream>>>(ffn_w1, ffn_b1, ffn_w2, ffn_b2,
                                                 out);
}